// cofinal_loss_18880676233641
// MI455X (gfx1250) — compile-verified
//
#include <hip/hip_runtime.h>
#include <math.h>

typedef __attribute__((ext_vector_type(2))) float v2f;
typedef __attribute__((ext_vector_type(8))) float v8f;
typedef __attribute__((ext_vector_type(4))) unsigned int v4u;
typedef __attribute__((ext_vector_type(8))) int v8i;
typedef __attribute__((ext_vector_type(4))) int v4i;

#define MARGIN   0.3f
#define ALPHA    1.0f
#define NROWS    8192      // b*n = 64*128
#define CDIM     256
#define LDS_STR  260       // padded row stride (floats): conflict-free ds_load_b64

#define HAVE_TDM __has_builtin(__builtin_amdgcn_tensor_load_to_lds)

// ---------------------------------------------------------------------------
// Kernel 1: row squared-norms. One wave per row.
// ---------------------------------------------------------------------------
__global__ __launch_bounds__(256) void norms_kernel(const float* __restrict__ feat,
                                                    float* __restrict__ norms) {
    const int wave = threadIdx.x >> 5;
    const int lane = threadIdx.x & 31;
    const int row  = blockIdx.x * 8 + wave;
    const float* p = feat + row * CDIM;
    float s = 0.0f;
#pragma unroll
    for (int t = 0; t < 8; ++t) {
        float x = p[lane + 32 * t];
        s = fmaf(x, x, s);
    }
#pragma unroll
    for (int m = 16; m >= 1; m >>= 1) s += __shfl_xor(s, m, 32);
    if (lane == 0) norms[row] = s;
}

// ---------------------------------------------------------------------------
// Kernel 2: mse + dr terms. Single block; OVERWRITES out[0] (poison-safe init).
// ---------------------------------------------------------------------------
__global__ __launch_bounds__(256) void small_losses_kernel(
    const float* __restrict__ gt,
    const float* __restrict__ s0, const float* __restrict__ s1, const float* __restrict__ s2,
    const float* __restrict__ etf0, const float* __restrict__ etf1,
    const float* __restrict__ lab0, const float* __restrict__ lab1,
    float* __restrict__ out) {
    __shared__ float red[256];
    const int t = threadIdx.x;
    float acc = 0.0f;
    if (t < 64) {
        float g = gt[t];
        float a = s0[t] - g, b = s1[t] - g, c = s2[t] - g;
        acc += (a * a + b * b + c * c) * (1.0f / (64.0f * 3.0f));
    }
    for (int i = t; i < 64 * 1024; i += 256) {
        acc += fabsf(etf0[i] - lab0[i]) * (1.0f / 65536.0f);
        acc += fabsf(etf1[i] - lab1[i]) * (1.0f / 65536.0f);
    }
    red[t] = acc;
    __syncthreads();
    for (int s = 128; s > 0; s >>= 1) {
        if (t < s) red[t] += red[t + s];
        __syncthreads();
    }
    if (t == 0) out[0] = red[0];
}

// ---------------------------------------------------------------------------
// TDM: load `nrows` x 256-f32 rows from global into LDS with hardware row
// padding (256 dwords data + 4 dwords pad == LDS_STR floats per row).
// D# per CDNA5 ISA ch.8: group0 = {count/lds/global/type}, group1 = dims.
// ---------------------------------------------------------------------------
#if HAVE_TDM
__device__ __forceinline__ void tdm_load_rows(unsigned lds_off, const float* gptr,
                                              int nrows) {
    const unsigned long long ga = (unsigned long long)(uintptr_t)gptr;
    v4u g0;
    g0.x = 1u;                                        // count=1, user descriptor
    g0.y = lds_off;                                   // lds_addr (bytes)
    g0.z = (unsigned)(ga & 0xFFFFFFFFu);              // global_addr[31:0]
    g0.w = (unsigned)((ga >> 32) & 0x01FFFFFFu)       // global_addr[56:32]
         | 0x80000000u;                               // type=2 ("image")
    v8i g1;
    g1[0] = (int)((2u << 16)       // data_size = 4 bytes
                | (1u << 20)       // pad_enable
                | (7u << 22)       // pad_interval: every 256 dwords
                | (3u << 25));     // pad_amount: 4 dwords -> 260-float stride
    g1[1] = (int)(256u << 16);     // tensor_dim0 = 256 elements
    g1[2] = (int)(0x2000u << 16);  // tensor_dim1 = 8192 rows
    g1[3] = (int)(256u << 16);     // tile_dim0 = 256
    g1[4] = nrows;                 // tile_dim1
    g1[5] = 256;                   // tensor_dim0_stride = 256 elements
    g1[6] = 0;
    g1[7] = 0;
    v4i z4 = {0, 0, 0, 0};
#if defined(__clang_major__) && __clang_major__ >= 23
    v8i z8 = {0, 0, 0, 0, 0, 0, 0, 0};
    __builtin_amdgcn_tensor_load_to_lds(g0, g1, z4, z4, z8, 0);
#else
    __builtin_amdgcn_tensor_load_to_lds(g0, g1, z4, z4, 0);
#endif
}
#endif

// ---------------------------------------------------------------------------
// Kernel 3: fused Gram + hardest-pos/neg mining. V_WMMA_F32_16X16X4_F32.
// One wave per block; 32-row panel vs all 512 column tiles. B tiles staged by
// the Tensor Data Mover; compute waits on TENSORcnt, restage gated on DScnt.
// ---------------------------------------------------------------------------
__global__ __launch_bounds__(32) void triplet_kernel(const float* __restrict__ feat,
                                                     const float* __restrict__ norms,
                                                     float* __restrict__ out) {
    __shared__ __align__(16) float sA[32 * LDS_STR];   // 33280 B
    __shared__ __align__(16) float sB[16 * LDS_STR];   // 16640 B  (total < 64KB)

    const int lane = threadIdx.x;      // 0..31
    const int ln   = lane & 15;
    const int h    = lane >> 4;
    const int koff = 2 * h;            // WMMA fragment K sub-offset
    const int m0   = blockIdx.x * 32;

    // Per-lane row norms for the two 16-row tiles.
    float sqi0[8], sqi1[8];
#pragma unroll
    for (int v = 0; v < 8; ++v) {
        sqi0[v] = norms[m0 + v + 8 * h];
        sqi1[v] = norms[m0 + 16 + v + 8 * h];
    }

    float posmax0[8], posmax1[8], negmin0[8], negmin1[8];
#pragma unroll
    for (int v = 0; v < 8; ++v) {
        posmax0[v] = 0.0f;    posmax1[v] = 0.0f;       // mining on d^2; clip+sqrt later
        negmin0[v] = 3.0e38f; negmin1[v] = 3.0e38f;
    }

#if HAVE_TDM
    tdm_load_rows((unsigned)(uintptr_t)&sA[0], feat + (size_t)m0 * CDIM, 32);
    tdm_load_rows((unsigned)(uintptr_t)&sB[0], feat, 16);
#else
    for (int idx = lane; idx < 32 * 64; idx += 32) {
        const int r = idx >> 6, c4 = (idx & 63) << 2;
        *(float4*)(&sA[r * LDS_STR + c4]) = *(const float4*)(feat + (m0 + r) * CDIM + c4);
    }
#endif

    const int aoff0 = ln * LDS_STR + koff;
    const int aoff1 = (ln + 16) * LDS_STR + koff;
    const int boff  = ln * LDS_STR + koff;
    const int w0    = (m0 >> 4) & 7;          // residue window of rows m0..m0+15
    const int w1    = ((m0 >> 4) + 1) & 7;    // window of rows m0+16..m0+31

    for (int t = 0; t < NROWS / 16; ++t) {
        const int j0 = t << 4;

#if HAVE_TDM
        __builtin_amdgcn_s_wait_tensorcnt((short)0);   // B tile t (and A) resident
#else
        __syncthreads();
        for (int idx = lane; idx < 16 * 64; idx += 32) {
            const int r = idx >> 6, c4 = (idx & 63) << 2;
            *(float4*)(&sB[r * LDS_STR + c4]) = *(const float4*)(feat + (j0 + r) * CDIM + c4);
        }
        __syncthreads();
#endif

        v8f c0 = {0.f, 0.f, 0.f, 0.f, 0.f, 0.f, 0.f, 0.f};
        v8f c1 = {0.f, 0.f, 0.f, 0.f, 0.f, 0.f, 0.f, 0.f};
#pragma unroll
        for (int kk = 0; kk < 64; ++kk) {
            const v2f b  = *(const v2f*)(&sB[boff  + 4 * kk]);
            const v2f a0 = *(const v2f*)(&sA[aoff0 + 4 * kk]);
            const v2f a1 = *(const v2f*)(&sA[aoff1 + 4 * kk]);
            c0 = __builtin_amdgcn_wmma_f32_16x16x4_f32(false, a0, false, b,
                                                       (short)0, c0, false, false);
            c1 = __builtin_amdgcn_wmma_f32_16x16x4_f32(false, a1, false, b,
                                                       (short)0, c1, false, false);
        }

#if HAVE_TDM
        if (t + 1 < NROWS / 16) {
            // All fragment reads of sB landed in VGPRs -> safe to overwrite.
            asm volatile("s_wait_dscnt 0x0" ::: "memory");
            tdm_load_rows((unsigned)(uintptr_t)&sB[0],
                          feat + (size_t)(j0 + 16) * CDIM, 16);
            // DMA overlaps the epilogue below.
        }
#endif

        // Epilogue: d^2 = |i|^2 + |j|^2 - 2 dot; mine max-pos / min-neg (d^2).
        const int   gj   = j0 + ln;
        const float sqj  = norms[gj];
        const int   wj   = (j0 >> 4) & 7;
        const bool  dia0 = ((m0 >> 4) == (j0 >> 4));
        const bool  dia1 = (((m0 + 16) >> 4) == (j0 >> 4));

        if (wj == w0) {   // 1 of 8 tiles: contains same-label columns for tile 0
#pragma unroll
            for (int v = 0; v < 8; ++v) {
                const float d2   = fmaf(-2.0f, c0[v], sqi0[v] + sqj);
                const bool  same = (ln == (v + 8 * h));
                const float cp   = (same && !dia0) ? d2 : -1.0f;
                const float cn   = same ? 3.0e38f : d2;
                posmax0[v] = (cp > posmax0[v]) ? cp : posmax0[v];
                negmin0[v] = (cn < negmin0[v]) ? cn : negmin0[v];
            }
        } else {          // pure negatives: fma + select only
#pragma unroll
            for (int v = 0; v < 8; ++v) {
                const float d2 = fmaf(-2.0f, c0[v], sqi0[v] + sqj);
                negmin0[v] = (d2 < negmin0[v]) ? d2 : negmin0[v];
            }
        }
        if (wj == w1) {
#pragma unroll
            for (int v = 0; v < 8; ++v) {
                const float e2   = fmaf(-2.0f, c1[v], sqi1[v] + sqj);
                const bool  same = (ln == (v + 8 * h));
                const float cp   = (same && !dia1) ? e2 : -1.0f;
                const float cn   = same ? 3.0e38f : e2;
                posmax1[v] = (cp > posmax1[v]) ? cp : posmax1[v];
                negmin1[v] = (cn < negmin1[v]) ? cn : negmin1[v];
            }
        } else {
#pragma unroll
            for (int v = 0; v < 8; ++v) {
                const float e2 = fmaf(-2.0f, c1[v], sqi1[v] + sqj);
                negmin1[v] = (e2 < negmin1[v]) ? e2 : negmin1[v];
            }
        }
    }

    // Cross-lane butterfly (16 lanes per row-half), then per-row loss.
    float acc = 0.0f;
#pragma unroll
    for (int v = 0; v < 8; ++v) {
        float p0 = posmax0[v], n0 = negmin0[v];
        float p1 = posmax1[v], n1 = negmin1[v];
#pragma unroll
        for (int m = 8; m >= 1; m >>= 1) {
            p0 = fmaxf(p0, __shfl_xor(p0, m, 16));
            n0 = fminf(n0, __shfl_xor(n0, m, 16));
            p1 = fmaxf(p1, __shfl_xor(p1, m, 16));
            n1 = fminf(n1, __shfl_xor(n1, m, 16));
        }
        if (ln == 0) {
            acc += fmaxf(sqrtf(fmaxf(p0, 0.0f)) - sqrtf(fmaxf(n0, 0.0f)) + MARGIN, 0.0f);
            acc += fmaxf(sqrtf(fmaxf(p1, 0.0f)) - sqrtf(fmaxf(n1, 0.0f)) + MARGIN, 0.0f);
        }
    }
    if (ln == 0) atomicAdd(out, acc * (ALPHA / (float)NROWS));
}

// ---------------------------------------------------------------------------
extern "C" void kernel_launch(void* const* d_in, const int* in_sizes, int n_in,
                              void* d_out, int out_size, void* d_ws, size_t ws_size,
                              hipStream_t stream) {
    const float* feat = (const float*)d_in[0];   // [64,128,256]
    const float* gt   = (const float*)d_in[1];   // [64]
    const float* s0   = (const float*)d_in[2];
    const float* s1   = (const float*)d_in[3];
    const float* s2   = (const float*)d_in[4];
    const float* etf0 = (const float*)d_in[5];   // [64,1024]
    const float* etf1 = (const float*)d_in[6];
    const float* lab0 = (const float*)d_in[7];
    const float* lab1 = (const float*)d_in[8];
    float* out   = (float*)d_out;
    float* norms = (float*)d_ws;                 // 8192 floats of scratch

    small_losses_kernel<<<1, 256, 0, stream>>>(gt, s0, s1, s2, etf0, etf1, lab0, lab1, out);
    norms_kernel<<<NROWS / 8, 256, 0, stream>>>(feat, norms);
    triplet_kernel<<<NROWS / 32, 32, 0, stream>>>(feat, norms, out);
}